// Quant_ReLU_55671366090843
// MI455X (gfx1250) — compile-verified
//
#include <hip/hip_runtime.h>

// ---------------------------------------------------------------------------
// Problem geometry (fixed by the reference): x is (256, 512, 1024) f32.
// Per-row min/max over 512*1024 elements, then elementwise quant/dequant/ReLU.
// ---------------------------------------------------------------------------
#define ROWS            256
#define ROWLEN          (512 * 1024)          // 524288 floats per row
#define RED_BPR         16                    // reduce blocks per row
#define RED_BLK_ELTS    (ROWLEN / RED_BPR)    // 32768 floats per reduce block
#define CHUNK_ELTS      4096                  // 16 KB TDM chunk
#define NCHUNK          (RED_BLK_ELTS / CHUNK_ELTS)  // 8
#define APPLY_BPR       128                   // apply blocks per row (4096 elts/block)

typedef __attribute__((ext_vector_type(4))) float        v4f;
typedef __attribute__((ext_vector_type(4))) unsigned int v4u;
typedef __attribute__((ext_vector_type(4))) int          v4i;
typedef __attribute__((ext_vector_type(8))) int          v8i;

// ---------------------------------------------------------------------------
// TDM: issue a 1-D tensor_load_to_lds of CHUNK_ELTS f32 from `src` into LDS
// byte-offset `lds_off`. Descriptor per CDNA5 ISA ch.8 (D# groups 0..3).
// This toolchain exposes the 6-arg builtin:
//   (uint32x4 g0, int32x8 g1, int32x4 g2, int32x4 g3, int32x8 extra, i32 cpol)
// ---------------------------------------------------------------------------
__device__ __forceinline__ void tdm_load_chunk(const float* src, unsigned lds_off) {
    unsigned long long ga = (unsigned long long)(const void*)src;

    v4u g0;
    g0.x = 1u;                                        // count=1, user mode, no gather
    g0.y = lds_off;                                   // lds_addr (bytes)
    g0.z = (unsigned)(ga & 0xFFFFFFFFu);              // global_addr[31:0]
    g0.w = (unsigned)((ga >> 32) & 0x01FFFFFFu)       // global_addr[56:32]
         | (2u << 30);                                // type=2 ("image")

    v8i g1;
    g1.s0 = 0x00020000;                               // wg_mask=0, data_size=2 (4B), no pad/iter
    g1.s1 = (int)((CHUNK_ELTS & 0xFFFFu) << 16);      // tensor_dim0[15:0] @bits63:48
    g1.s2 = (int)(((CHUNK_ELTS >> 16) & 0xFFFFu)      // tensor_dim0[31:16]
         | (1u << 16));                               // tensor_dim1 = 1
    g1.s3 = (int)((CHUNK_ELTS & 0xFFFFu) << 16);      // tile_dim0 @bits127:112
    g1.s4 = 1;                                        // tile_dim1 = 1, tile_dim2 = 0
    g1.s5 = CHUNK_ELTS;                               // tensor_dim0_stride[31:0]
    g1.s6 = 0;                                        // stride hi / dim1_stride lo
    g1.s7 = 0;

    v4i g2 = {0, 0, 0, 0};                            // dims 2/3 unused
    v4i g3 = {0, 0, 0, 0};
    v8i gx = {0, 0, 0, 0, 0, 0, 0, 0};                // 6-arg form extra operand

    __builtin_amdgcn_tensor_load_to_lds(g0, g1, g2, g3, gx, /*cpol=*/0);
}

__device__ __forceinline__ float qdq_relu(float v, float m, float rng) {
    // Bit-faithful to the reference sequence:
    //   q   = round(254*(x-m)/rng - 127) / 1000      (round = RNE, like jnp.round)
    //   deq = (1000*q + 127) * rng / 254 + m
    //   out = relu(deq)
    float t = 254.0f * (v - m) / rng;
    float q = __builtin_rintf(t - 127.0f) / 1000.0f;
    float d = (1000.0f * q + 127.0f) * rng / 254.0f + m;
    return fmaxf(d, 0.0f);
}

// ---------------------------------------------------------------------------
// Kernel 1: per-block min/max via TDM double-buffered streaming into LDS.
// grid = ROWS * RED_BPR blocks, 256 threads (8 waves).
// ---------------------------------------------------------------------------
__global__ __launch_bounds__(256) void quant_reduce_kernel(
    const float* __restrict__ x, float* __restrict__ pmin, float* __restrict__ pmax) {
    __shared__ __align__(16) float buf[2][CHUNK_ELTS];   // 2 x 16 KB ping/pong
    __shared__ float smin[256];
    __shared__ float smax[256];

    const int t   = threadIdx.x;
    const int row = blockIdx.x >> 4;          // /RED_BPR
    const int blk = blockIdx.x & (RED_BPR - 1);
    const float* base = x + (size_t)row * ROWLEN + (size_t)blk * RED_BLK_ELTS;

    const unsigned lds0 = (unsigned)(unsigned long long)(void*)&buf[0][0];
    const unsigned lds1 = (unsigned)(unsigned long long)(void*)&buf[1][0];

    float vmin =  __builtin_inff();
    float vmax = -__builtin_inff();

    // Wave 0 drives the TDM pipeline (TDM ignores EXEC; uniform branch keeps
    // waves 1..7 from duplicating the DMA).
    if (t < 32) {
        tdm_load_chunk(base, lds0);
    }

    for (int c = 0; c < NCHUNK; ++c) {
        if (t < 32) {
            if (c + 1 < NCHUNK) {
                tdm_load_chunk(base + (size_t)(c + 1) * CHUNK_ELTS,
                               ((c + 1) & 1) ? lds1 : lds0);
                __builtin_amdgcn_s_wait_tensorcnt(1);   // chunk c landed
            } else {
                __builtin_amdgcn_s_wait_tensorcnt(0);   // last chunk landed
            }
        }
        __syncthreads();                                // publish buf[c&1]

        const v4f* b4 = (const v4f*)buf[c & 1];
        #pragma unroll
        for (int i = 0; i < CHUNK_ELTS / 4 / 256; ++i) { // 4 x ds_load_b128
            v4f v = b4[t + i * 256];
            vmin = fminf(vmin, fminf(fminf(v.x, v.y), fminf(v.z, v.w)));
            vmax = fmaxf(vmax, fmaxf(fmaxf(v.x, v.y), fmaxf(v.z, v.w)));
        }
        __syncthreads();                                // safe to overwrite buffer
    }

    smin[t] = vmin;
    smax[t] = vmax;
    __syncthreads();
    for (int s = 128; s > 0; s >>= 1) {
        if (t < s) {
            smin[t] = fminf(smin[t], smin[t + s]);
            smax[t] = fmaxf(smax[t], smax[t + s]);
        }
        __syncthreads();
    }
    if (t == 0) {
        pmin[blockIdx.x] = smin[0];
        pmax[blockIdx.x] = smax[0];
    }
}

// ---------------------------------------------------------------------------
// Kernel 2: fold RED_BPR partials per row. 1 block, 256 threads (1 row/thread).
// ---------------------------------------------------------------------------
__global__ __launch_bounds__(256) void quant_finalize_kernel(
    const float* __restrict__ pmin, const float* __restrict__ pmax,
    float* __restrict__ rmin, float* __restrict__ rmax) {
    const int r = threadIdx.x;
    float m =  __builtin_inff();
    float M = -__builtin_inff();
    #pragma unroll
    for (int i = 0; i < RED_BPR; ++i) {
        m = fminf(m, pmin[r * RED_BPR + i]);
        M = fmaxf(M, pmax[r * RED_BPR + i]);
    }
    rmin[r] = m;
    rmax[r] = M;
}

// ---------------------------------------------------------------------------
// Kernel 3: elementwise quant/dequant/ReLU, non-temporal streaming.
// grid = ROWS * APPLY_BPR blocks, 256 threads, 4 x float4 per thread.
// ---------------------------------------------------------------------------
__global__ __launch_bounds__(256) void quant_apply_kernel(
    const float* __restrict__ x, const float* __restrict__ rmin,
    const float* __restrict__ rmax, float* __restrict__ out) {
    const int t   = threadIdx.x;
    const int row = blockIdx.x >> 7;                 // /APPLY_BPR
    const int blk = blockIdx.x & (APPLY_BPR - 1);

    const float m   = rmin[row];                     // row-uniform -> s_load
    const float rng = rmax[row] - m;

    const size_t base4 = ((size_t)row * ROWLEN + (size_t)blk * 4096) >> 2;
    const v4f* x4 = (const v4f*)x;
    v4f*       o4 = (v4f*)out;

    #pragma unroll
    for (int i = 0; i < 4; ++i) {
        size_t idx = base4 + (size_t)t + (size_t)i * 256;
        v4f v = __builtin_nontemporal_load(&x4[idx]);  // th:NT, no reuse
        v4f r;
        r.x = qdq_relu(v.x, m, rng);
        r.y = qdq_relu(v.y, m, rng);
        r.z = qdq_relu(v.z, m, rng);
        r.w = qdq_relu(v.w, m, rng);
        __builtin_nontemporal_store(r, &o4[idx]);      // streaming store
    }
}

// ---------------------------------------------------------------------------
// Host-side launch. Workspace layout (floats):
//   [0, 4096)      pmin partials
//   [4096, 8192)   pmax partials
//   [8192, 8448)   rmin
//   [8448, 8704)   rmax
// ---------------------------------------------------------------------------
extern "C" void kernel_launch(void* const* d_in, const int* in_sizes, int n_in,
                              void* d_out, int out_size, void* d_ws, size_t ws_size,
                              hipStream_t stream) {
    (void)in_sizes; (void)n_in; (void)out_size; (void)ws_size;
    const float* x = (const float*)d_in[0];
    float* out  = (float*)d_out;
    float* pmin = (float*)d_ws;
    float* pmax = pmin + ROWS * RED_BPR;
    float* rmin = pmax + ROWS * RED_BPR;
    float* rmax = rmin + ROWS;

    quant_reduce_kernel<<<ROWS * RED_BPR, 256, 0, stream>>>(x, pmin, pmax);
    quant_finalize_kernel<<<1, 256, 0, stream>>>(pmin, pmax, rmin, rmax);
    quant_apply_kernel<<<ROWS * APPLY_BPR, 256, 0, stream>>>(x, rmin, rmax, out);
}